// BoundaryLoss_25640954757313
// MI455X (gfx1250) — compile-verified
//
#include <hip/hip_runtime.h>

typedef __attribute__((ext_vector_type(2))) float v2f;
typedef __attribute__((ext_vector_type(8))) float v8f;

static constexpr int NB = 16;   // batch
static constexpr int NH = 512;  // rows
static constexpr int NW = 512;  // cols

// ---------------------------------------------------------------------------
// Kernel A: exact 1D column pass.
// One thread per (b, j) column. Forward scan tracks nearest foreground row at
// or above (sentinel -2048 = -2*(H+W)); backward scan tracks nearest at or
// below (sentinel 2560 = 2*(H+W)+H). Matches reference cummax/cummin exactly.
// d1[b,i,j] = min((i-f)^2, (b-i)^2). Coalesced: adjacent threads = adjacent j.
// ---------------------------------------------------------------------------
__global__ void edt_col_pass(const int* __restrict__ gt, float* __restrict__ d1)
{
    int tid = blockIdx.x * blockDim.x + threadIdx.x;   // 0 .. B*W-1
    int b = tid >> 9;
    int j = tid & (NW - 1);
    const int*  g = gt + (size_t)b * NH * NW + j;
    float*      d = d1 + (size_t)b * NH * NW + j;

    const float BIGF = 2048.0f;                        // 2*(H+W)
    float f = -BIGF;
    for (int i = 0; i < NH; ++i) {
        if (g[(size_t)i * NW] != 0) f = (float)i;
        float di = (float)i - f;
        d[(size_t)i * NW] = di * di;
    }
    float bb = BIGF + (float)NH;                       // 2560
    for (int i = NH - 1; i >= 0; --i) {
        if (g[(size_t)i * NW] != 0) bb = (float)i;
        float di = bb - (float)i;
        float fwd = d[(size_t)i * NW];
        d[(size_t)i * NW] = fminf(fwd, di * di);
    }
}

// ---------------------------------------------------------------------------
// Kernel B: exact row min-plus + weighted partial sum.
// One 256-thread workgroup per (b, i) row. d1 row staged in LDS; each thread
// owns outputs j and j+256 and scans all 512 parabola bases (broadcast LDS
// reads). Per-row sum of probs*sqrt(d2) is reduced with V_WMMA_F32_16X16X4_F32
// using A = ones(16x4): every row of D holds the column sums of B, so the
// wave total = sum over lanes 0..15 of D vgpr0 (M=0 row) -- exact f32 math.
// ---------------------------------------------------------------------------
__global__ void edt_row_loss(const float* __restrict__ d1,
                             const float* __restrict__ probs,
                             float* __restrict__ partials)
{
    __shared__ float s_d1[NW];
    __shared__ float s_wsum[8];

    int row = blockIdx.x;            // b*H + i
    int b   = row >> 9;
    int i   = row & (NH - 1);
    int t   = threadIdx.x;           // 0..255

    const float* drow = d1 + (size_t)row * NW;
    // probs layout [B, 2, H, W], channel 0
    const float* prow = probs + (((size_t)b * 2) * NH + i) * NW;

    // Stage the d1 row into LDS (coalesced 8B loads, ds stores).
    float2 v = reinterpret_cast<const float2*>(drow)[t];
    s_d1[2 * t]     = v.x;
    s_d1[2 * t + 1] = v.y;
    // Warm L2/L0 for the probs row we need after the compute loop.
    __builtin_prefetch(&prow[t],       0, 1);
    __builtin_prefetch(&prow[t + 256], 0, 1);
    __syncthreads();

    float fj0 = (float)t;
    float fj1 = (float)(t + 256);
    float m0 = 3.0e38f, m1 = 3.0e38f;
    #pragma unroll 8
    for (int jp = 0; jp < NW; ++jp) {
        float dv = s_d1[jp];
        float fp = (float)jp;
        float a0 = fj0 - fp;
        float a1 = fj1 - fp;
        m0 = fminf(m0, fmaf(a0, a0, dv));
        m1 = fminf(m1, fmaf(a1, a1, dv));
    }

    float s0 = prow[t]       * sqrtf(m0);
    float s1 = prow[t + 256] * sqrtf(m1);

    // WMMA wave reduction: D = ones(16x4) x B(4x16) + 0  => rows of D are
    // column sums of B; the 64 per-wave values are exactly B's elements.
    v2f amat; amat[0] = 1.0f; amat[1] = 1.0f;
    v2f bmat; bmat[0] = s0;   bmat[1] = s1;
    v8f cz = {};
    v8f dacc = __builtin_amdgcn_wmma_f32_16x16x4_f32(
        /*neg_a=*/false, amat, /*neg_b=*/false, bmat,
        /*c_mod=*/(short)0, cz, /*reuse_a=*/false, /*reuse_b=*/false);

    // Sum the 16 column sums held in lanes 0..15 of dacc[0] (M=0 row).
    float r = dacc[0];
    r += __shfl_xor(r, 1);
    r += __shfl_xor(r, 2);
    r += __shfl_xor(r, 4);
    r += __shfl_xor(r, 8);

    int lane = t & 31;
    int wave = t >> 5;                 // 8 wave32's per block
    if (lane == 0) s_wsum[wave] = r;
    __syncthreads();
    if (t == 0) {
        float tot = 0.0f;
        #pragma unroll
        for (int w = 0; w < 8; ++w) tot += s_wsum[w];
        partials[row] = tot;
    }
}

// ---------------------------------------------------------------------------
// Kernel C: deterministic final reduction of B*H = 8192 row partials.
// Fixed-order per-thread accumulation + LDS tree; writes mean to d_out[0].
// ---------------------------------------------------------------------------
__global__ void edt_final_reduce(const float* __restrict__ partials,
                                 float* __restrict__ out)
{
    __shared__ float s[256];
    int t = threadIdx.x;
    float acc = 0.0f;
    for (int k = t; k < NB * NH; k += 256) acc += partials[k];
    s[t] = acc;
    __syncthreads();
    for (int off = 128; off > 0; off >>= 1) {
        if (t < off) s[t] += s[t + off];
        __syncthreads();
    }
    if (t == 0) out[0] = s[0] * (1.0f / (float)((size_t)NB * NH * NW));
}

extern "C" void kernel_launch(void* const* d_in, const int* in_sizes, int n_in,
                              void* d_out, int out_size, void* d_ws, size_t ws_size,
                              hipStream_t stream)
{
    (void)in_sizes; (void)n_in; (void)out_size; (void)ws_size;
    const float* probs = (const float*)d_in[0];   // [16, 2, 512, 512] f32
    const int*   gt    = (const int*)d_in[1];     // [16, 1, 512, 512] i32

    float* d1       = (float*)d_ws;                                    // 16.78 MB
    float* partials = (float*)((char*)d_ws +
                               (size_t)NB * NH * NW * sizeof(float));  // 32 KB

    edt_col_pass    <<<(NB * NW) / 256, 256, 0, stream>>>(gt, d1);
    edt_row_loss    <<<NB * NH,         256, 0, stream>>>(d1, probs, partials);
    edt_final_reduce<<<1,               256, 0, stream>>>(partials, (float*)d_out);
}